// HNHNConv_37254546325797
// MI455X (gfx1250) — compile-verified
//
#include <hip/hip_runtime.h>
#include <hip/hip_bf16.h>

typedef float v2f __attribute__((ext_vector_type(2)));
typedef float v8f __attribute__((ext_vector_type(8)));

#define C_DIM 128
#define LDS_STRIDE 132   // 128 + 4 pad -> conflict-free column reads
#define N_EDGES_CONST 25000

// -------------------------------------------------------------------------
// out[M, 128] = in[M, 128] @ W[128,128]^T + bias   (W row-major [oc][ic])
// 1 block = 16 rows; 8 waves, wave w computes column tile [16w, 16w+16).
// A-slab (16x128 f32) staged in LDS, shared by all 8 waves.
// WMMA f32 16x16x4: A lane<16 holds {K=k,k+1} of row M=lane; lane>=16 holds
// {K=k+2,k+3} of row M=lane-16.  B mirrored over N.  D: vgpr v, lane<16 ->
// (M=v, N=lane), lane>=16 -> (M=8+v, N=lane-16).
// -------------------------------------------------------------------------
__global__ __launch_bounds__(256) void hnhn_gemm_bias(
    const float* __restrict__ in, const float* __restrict__ W,
    const float* __restrict__ bias, float* __restrict__ out, int M)
{
    __shared__ float lds[16 * LDS_STRIDE];
    const int tid = threadIdx.x;
    const int m0  = blockIdx.x * 16;

    // cooperative coalesced load of the 16x128 A slab (zero-pad past M)
#pragma unroll
    for (int i = 0; i < 8; ++i) {
        int idx = tid + i * 256;       // 0..2047
        int r = idx >> 7;
        int c = idx & 127;
        int gr = m0 + r;
        lds[r * LDS_STRIDE + c] = (gr < M) ? in[gr * C_DIM + c] : 0.0f;
    }
    __syncthreads();

    const int wave  = tid >> 5;
    const int lane  = tid & 31;
    const int n0    = wave * 16;
    const int l15   = lane & 15;
    const int khalf = (lane >> 4) << 1;          // 0 or 2

    const float* ldsA = &lds[l15 * LDS_STRIDE + khalf];
    const float* gB   = &W[(n0 + l15) * C_DIM + khalf];

    v8f acc = {};
#if __has_builtin(__builtin_amdgcn_wmma_f32_16x16x4_f32)
#pragma unroll 8
    for (int k = 0; k < C_DIM; k += 4) {
        v2f a; a.x = ldsA[k]; a.y = ldsA[k + 1];
        v2f b; b.x = gB[k];   b.y = gB[k + 1];
        acc = __builtin_amdgcn_wmma_f32_16x16x4_f32(
            false, a, false, b, (short)0, acc, false, false);
    }
#else
    // scalar fallback reproducing the D layout
    {
        const int colf  = n0 + l15;
        const int rloc  = (lane >> 4) << 3;
#pragma unroll
        for (int v = 0; v < 8; ++v) {
            float s = 0.0f;
            for (int k = 0; k < C_DIM; ++k)
                s = fmaf(lds[(rloc + v) * LDS_STRIDE + k], W[colf * C_DIM + k], s);
            acc[v] = s;
        }
    }
#endif

    const int col   = n0 + l15;
    const float bv  = bias[col];
    const int rbase = m0 + ((lane >> 4) << 3);
    float* op = &out[(size_t)rbase * C_DIM + col];

    if (m0 + 16 <= M) {
        // full tile (the overwhelmingly common case): unconditional stores
#pragma unroll
        for (int v = 0; v < 8; ++v)
            op[v * C_DIM] = acc[v] + bv;
    } else {
        // M-tail tile: per-row guard
#pragma unroll
        for (int v = 0; v < 8; ++v)
            if (rbase + v < M) op[v * C_DIM] = acc[v] + bv;
    }
}

// -------------------------------------------------------------------------
// One wave per incidence: gather 128 f32 (float4/lane) from src[src_idx],
// atomic-add into acc[dst_idx]; lane 0 bumps the degree counter.
// -------------------------------------------------------------------------
__global__ __launch_bounds__(256) void hnhn_scatter(
    const float* __restrict__ src, const int* __restrict__ src_idx,
    const int* __restrict__ dst_idx, float* __restrict__ acc,
    float* __restrict__ cnt, int n_inc)
{
    int t    = blockIdx.x * 256 + threadIdx.x;
    int inc  = t >> 5;
    int lane = t & 31;
    if (inc >= n_inc) return;
    int s = src_idx[inc];
    int d = dst_idx[inc];
    if (lane == 0) atomicAdd(&cnt[d], 1.0f);
    const float4 v = *(const float4*)&src[s * C_DIM + lane * 4];
    float* a = &acc[d * C_DIM + lane * 4];
    atomicAdd(a + 0, v.x);
    atomicAdd(a + 1, v.y);
    atomicAdd(a + 2, v.z);
    atomicAdd(a + 3, v.w);
}

// out[t] = relu(acc[t] / max(cnt[row], 1))   (acc may alias out)
__global__ __launch_bounds__(256) void hnhn_finalize(
    const float* __restrict__ acc, const float* __restrict__ cnt,
    float* __restrict__ out, int n_rows)
{
    int t = blockIdx.x * 256 + threadIdx.x;
    if (t >= n_rows * C_DIM) return;
    int row = t >> 7;
    float inv = 1.0f / fmaxf(cnt[row], 1.0f);
    out[t] = fmaxf(acc[t] * inv, 0.0f);
}

__global__ __launch_bounds__(256) void hnhn_zero4(float4* __restrict__ p, int n4)
{
    int t = blockIdx.x * 256 + threadIdx.x;
    if (t < n4) p[t] = make_float4(0.f, 0.f, 0.f, 0.f);
}

static inline int cdiv(int a, int b) { return (a + b - 1) / b; }

extern "C" void kernel_launch(void* const* d_in, const int* in_sizes, int n_in,
                              void* d_out, int out_size, void* d_ws, size_t ws_size,
                              hipStream_t stream)
{
    const float* x        = (const float*)d_in[0];
    const int*   hidx     = (const int*)d_in[1];
    const float* W_v2e    = (const float*)d_in[2];
    const float* b_v2e    = (const float*)d_in[3];
    const float* W_e2v    = (const float*)d_in[4];
    const float* b_e2v    = (const float*)d_in[5];
    float*       out      = (float*)d_out;

    const int C     = in_sizes[3];            // 128
    const int N     = in_sizes[0] / C;        // 50000 nodes
    const int N_INC = in_sizes[1] / 2;        // 600000 incidences
    const int M     = N_EDGES_CONST;          // 25000 hyperedges

    const int* node_idx = hidx;               // hyperedge_index[0]
    const int* edge_idx = hidx + N_INC;       // hyperedge_index[1]

    // workspace layout (floats)
    float* ws    = (float*)d_ws;
    float* h     = ws;                        // N*C   (node projections)
    float* e_acc = h + (size_t)N * C;         // M*C   (edge accum, reused as GEMM2 out)
    float* e     = e_acc + (size_t)M * C;     // M*C   (edge features post mean+relu)
    float* cnt_e = e + (size_t)M * C;         // M
    float* cnt_n = cnt_e + M;                 // N

    // 1. zero edge accum + both counters (contiguous region), and d_out accum
    {
        int z1   = 2 * M * C + M + N;         // e_acc + e + cnt_e + cnt_n
        int z1_4 = z1 / 4;
        hnhn_zero4<<<cdiv(z1_4, 256), 256, 0, stream>>>((float4*)e_acc, z1_4);
        int z2_4 = (N * C) / 4;
        hnhn_zero4<<<cdiv(z2_4, 256), 256, 0, stream>>>((float4*)out, z2_4);
    }

    // 2. h = x @ W_v2e^T + b_v2e
    hnhn_gemm_bias<<<cdiv(N, 16), 256, 0, stream>>>(x, W_v2e, b_v2e, h, N);

    // 3. scatter-sum node projections into edges (+ edge degrees)
    hnhn_scatter<<<cdiv(N_INC * 32, 256), 256, 0, stream>>>(
        h, node_idx, edge_idx, e_acc, cnt_e, N_INC);

    // 4. e = relu(mean)
    hnhn_finalize<<<cdiv(M * C, 256), 256, 0, stream>>>(e_acc, cnt_e, e, M);

    // 5. f = e @ W_e2v^T + b_e2v   (writes into e_acc, now free)
    hnhn_gemm_bias<<<cdiv(M, 16), 256, 0, stream>>>(e, W_e2v, b_e2v, e_acc, M);

    // 6. scatter-sum edge features into nodes (accumulate directly in d_out)
    hnhn_scatter<<<cdiv(N_INC * 32, 256), 256, 0, stream>>>(
        e_acc, edge_idx, node_idx, out, cnt_n, N_INC);

    // 7. out = relu(mean), in place
    hnhn_finalize<<<cdiv(N * C, 256), 256, 0, stream>>>(out, cnt_n, out, N);
}